// LineSearchStrongWolfeConditions_7739531067949
// MI455X (gfx1250) — compile-verified
//
#include <hip/hip_runtime.h>

typedef float v2f __attribute__((ext_vector_type(2)));
typedef float v8f __attribute__((ext_vector_type(8)));
typedef int   v4i __attribute__((vector_size(16)));   // GCC-style, matches builtin

#define WOLFE_C1 1e-4f
#define WOLFE_C2 0.9f
#define MAX_STEP 64.0f
#define WIDEN_ITERS 7
#define ZOOM_ITERS 20

#define D_DIM 2048
#define ROWS  1024        // B*N = 32*32

// workspace layout (float offsets)
#define OFF_XY0   0              // [16][1024] partial sums  x0 . (x0@A)
#define OFF_SDY0  (16 * 1024)    // [16][1024] partial sums  sd . (x0@A)
#define OFF_H     (32 * 1024)    // [16][1024] partial sums  sd . (sd@A)
#define OFF_E0    (48 * 1024)    // [1024]
#define OFF_G0    (49 * 1024)    // [1024]
#define OFF_HH    (50 * 1024)    // [1024]

// LDS staging for the B (A-matrix) panel
#define KB 32                    // K-block staged per buffer
#define RS 136                   // padded LDS row stride in floats (bank-group split)
#define NKB (D_DIM / KB)         // 64 K-blocks

#if __has_builtin(__builtin_amdgcn_global_load_async_to_lds_b128)
#define USE_ASYNC_LDS 1
#endif

// exact parameter types for the async builtin:
//   param0: v4i __device__ *  (addrspace(1), non-const)
//   param1: v4i __shared__ *  (addrspace(3))
typedef __attribute__((address_space(1))) v4i* gv4i_p;
typedef __attribute__((address_space(3))) v4i* lv4i_p;
#define AS_GLOBAL_V4I(p) ((gv4i_p)(unsigned long long)(p))
#define AS_LDS_V4I(p)    ((lv4i_p)(unsigned int)(unsigned long long)(p))

__device__ __forceinline__ void wait_async_zero() {
#ifdef USE_ASYNC_LDS
#if __has_builtin(__builtin_amdgcn_s_wait_asynccnt)
  __builtin_amdgcn_s_wait_asynccnt(0);
#else
  asm volatile("s_wait_asynccnt 0" ::: "memory");
#endif
#endif
}

// ---------------------------------------------------------------------------
// Kernel 1: fused fp32 WMMA GEMM + weighted row reductions.
// Logical GEMM: Y = X @ A,  X = [x0 rows (0..1023) ; sd rows (1024..2047)].
// Grid (nblk=16, mblk=16), block tile 128(M) x 128(N), 8 waves; each wave owns
// a 16(M) x 128(N) strip (rows never cross waves -> wave-local row sums).
// The B k-slab (KB x 128) is staged in LDS once per K-block (async copy with
// double buffering when available), removing the 8x cross-wave duplication of
// B-fragment global loads.  A-fragments stream directly from global (unique
// rows per wave).  K loop: V_WMMA_F32_16X16X4_F32 accumulation.
// ---------------------------------------------------------------------------
__global__ __launch_bounds__(256) void ls_gemm_kernel(
    const float* __restrict__ x0, const float* __restrict__ Amat,
    const float* __restrict__ sd, float* __restrict__ ws)
{
  __shared__ float lbuf[2][KB * RS];   // 2 x 17408 B

  const int t       = threadIdx.x;
  const int lane    = t & 31;
  const int lane_lo = lane & 15;
  const int lane_hi = lane >> 4;              // 0 or 1
  const int wave    = t >> 5;                 // 0..7
  const int nblk    = blockIdx.x;             // 0..15
  const int mblk    = blockIdx.y;             // 0..15
  const bool top    = (mblk < 8);             // top: x0 rows; bottom: sd rows
  const int rowBase = (top ? mblk : (mblk - 8)) * 128 + wave * 16;
  const float* __restrict__ X = top ? x0 : sd;
  const int n0 = nblk * 128;

  const v8f vzero = {0.f, 0.f, 0.f, 0.f, 0.f, 0.f, 0.f, 0.f};
  v8f acc[8];
#pragma unroll
  for (int j = 0; j < 8; ++j) acc[j] = vzero;

  // stage B rows [k0, k0+KB) x cols [n0, n0+128) into lbuf[sel]
  // KB*128/4 = 1024 b128 transfers, 4 per thread
  auto issue_fill = [&](int sel, int k0) {
#pragma unroll
    for (int r = 0; r < 4; ++r) {
      const int e   = t + 256 * r;            // 0..1023
      const int kk  = e >> 5;                 // 0..31
      const int col = (e & 31) * 4;           // 0..124
      const float* g = Amat + (size_t)(k0 + kk) * D_DIM + n0 + col;
      float* l = &lbuf[sel][kk * RS + col];
#ifdef USE_ASYNC_LDS
      __builtin_amdgcn_global_load_async_to_lds_b128(AS_GLOBAL_V4I(g), AS_LDS_V4I(l), 0, 0);
#else
      *reinterpret_cast<float4*>(l) = *reinterpret_cast<const float4*>(g);
#endif
    }
  };

  // A-fragment (16x4 f32): lane holds row M=lane_lo, K = vgpr + 2*lane_hi
  const float* aPtr = X + (size_t)(rowBase + lane_lo) * D_DIM + 2 * lane_hi;

  issue_fill(0, 0);

  for (int kb = 0; kb < NKB; ++kb) {
    wait_async_zero();        // this wave's copies into lbuf[kb&1] are done
    __syncthreads();          // all waves' copies done; prev reads finished
    if (kb + 1 < NKB) issue_fill((kb + 1) & 1, (kb + 1) * KB);

    // B-fragment (4x16 f32): lane holds col N=lane_lo, K = vgpr + 2*lane_hi
    const float* lrow = &lbuf[kb & 1][(2 * lane_hi) * RS + lane_lo];
    for (int kk = 0; kk < KB; kk += 4) {
      const v2f af = *reinterpret_cast<const v2f*>(aPtr);  // K, K+1 (b64)
      __builtin_prefetch(aPtr + 64, 0, 3);                 // near-scope prefetch
#pragma unroll
      for (int j = 0; j < 8; ++j) {
        v2f bf;
        bf.x = lrow[j * 16];
        bf.y = lrow[j * 16 + RS];
        acc[j] = __builtin_amdgcn_wmma_f32_16x16x4_f32(
            false, af, false, bf, (short)0, acc[j], false, false);
      }
      aPtr += 4;
      lrow += 4 * RS;
    }
  }

  // Epilogue: weighted row reductions.
  // C/D layout: M = vgpr + 8*lane_hi, N = lane_lo (per 16x16 tile j).
  if (top) {
#pragma unroll
    for (int v = 0; v < 8; ++v) {
      const int i = rowBase + v + 8 * lane_hi;           // global row 0..1023
      float sxy = 0.f, ssd = 0.f;
#pragma unroll
      for (int j = 0; j < 8; ++j) {
        const int n = n0 + j * 16 + lane_lo;
        const float y = acc[j][v];
        sxy = __builtin_fmaf(y, x0[(size_t)i * D_DIM + n], sxy);
        ssd = __builtin_fmaf(y, sd[(size_t)i * D_DIM + n], ssd);
      }
#pragma unroll
      for (int m = 1; m < 16; m <<= 1) {
        sxy += __shfl_xor(sxy, m, 32);
        ssd += __shfl_xor(ssd, m, 32);
      }
      if (lane_lo == 0) {
        ws[OFF_XY0  + nblk * ROWS + i] = sxy;
        ws[OFF_SDY0 + nblk * ROWS + i] = ssd;
      }
    }
  } else {
#pragma unroll
    for (int v = 0; v < 8; ++v) {
      const int i = rowBase + v + 8 * lane_hi;
      float sh = 0.f;
#pragma unroll
      for (int j = 0; j < 8; ++j) {
        const int n = n0 + j * 16 + lane_lo;
        sh = __builtin_fmaf(acc[j][v], sd[(size_t)i * D_DIM + n], sh);
      }
#pragma unroll
      for (int m = 1; m < 16; m <<= 1) sh += __shfl_xor(sh, m, 32);
      if (lane_lo == 0) ws[OFF_H + nblk * ROWS + i] = sh;
    }
  }
}

// ---------------------------------------------------------------------------
// Kernel 2: per-row scalars.  One workgroup (256 threads) per row.
//   E0 = 0.5*sum(x0 . Y0) - b.x0 ; g0 = sum(sd . Y0) - b.sd ; h = sum(sd . Y1)
// Partial sums over N-blocks folded in fixed order (deterministic).
// ---------------------------------------------------------------------------
__global__ __launch_bounds__(256) void ls_rowstats_kernel(
    const float* __restrict__ x0, const float* __restrict__ b,
    const float* __restrict__ sd, float* __restrict__ ws)
{
  const int i = blockIdx.x;
  float xb = 0.f, bsd = 0.f;
  for (int n = threadIdx.x; n < D_DIM; n += 256) {
    const float bb = b[n];
    xb  = __builtin_fmaf(x0[(size_t)i * D_DIM + n], bb, xb);
    bsd = __builtin_fmaf(sd[(size_t)i * D_DIM + n], bb, bsd);
  }
#pragma unroll
  for (int m = 1; m < 32; m <<= 1) {
    xb  += __shfl_xor(xb, m, 32);
    bsd += __shfl_xor(bsd, m, 32);
  }
  __shared__ float s0[8], s1[8];
  if ((threadIdx.x & 31) == 0) {
    s0[threadIdx.x >> 5] = xb;
    s1[threadIdx.x >> 5] = bsd;
  }
  __syncthreads();
  if (threadIdx.x == 0) {
    float txb = 0.f, tbsd = 0.f;
    for (int w = 0; w < 8; ++w) { txb += s0[w]; tbsd += s1[w]; }
    float sxy = 0.f, ssd = 0.f, sh = 0.f;
    for (int nb = 0; nb < 16; ++nb) {
      sxy += ws[OFF_XY0  + nb * ROWS + i];
      ssd += ws[OFF_SDY0 + nb * ROWS + i];
      sh  += ws[OFF_H    + nb * ROWS + i];
    }
    ws[OFF_E0 + i] = 0.5f * sxy - txb;
    ws[OFF_G0 + i] = ssd - tbsd;
    ws[OFF_HH + i] = sh;
  }
}

// ---------------------------------------------------------------------------
// Kernel 3: exact per-lane replay of widen (alg 3.5) + zoom (alg 3.6) on the
// 1-D quadratic E(t) = E0 + t*g0 + 0.5*t^2*h, G(t) = g0 + t*h, then write
// upper_x = x0 + t_up * sd.  One workgroup per row.
// ---------------------------------------------------------------------------
__global__ __launch_bounds__(256) void ls_wolfe_out_kernel(
    const float* __restrict__ x0, const float* __restrict__ sd,
    const float* __restrict__ ws, float* __restrict__ out)
{
  const int i = blockIdx.x;
  const float E0 = ws[OFF_E0 + i];
  const float g0 = ws[OFF_G0 + i];
  const float h  = ws[OFF_HH + i];

  float la = 0.f, ua = 1.f;         // bracket step params
  float t_lo = 0.f, t_up = 0.f;     // params of lower_x / upper_x
  float prev_error = E0;
  bool widening = true, zooming = false;

  // widening phase
  for (int idx = 0; idx < WIDEN_ITERS; ++idx) {
    if (widening) t_up = ua;                       // upper_x = x0 + ua*sd
    const float err = E0 + t_up * g0 + 0.5f * t_up * t_up * h;
    bool inc = widening && (err > E0 + WOLFE_C1 * ua * g0);
    if (idx > 0) inc = inc || (widening && (err >= prev_error));
    zooming = zooming || inc;
    widening = widening && !inc;
    const float gdot = g0 + t_up * h;
    const bool met = widening && (fabsf(gdot) <= -WOLFE_C2 * g0);
    widening = widening && !met;
    const bool swapping = widening && (gdot >= 0.f);
    zooming = zooming || swapping;
    if (swapping) {
      float tt = ua; ua = la; la = tt;
      tt = t_up; t_up = t_lo; t_lo = tt;
    }
    widening = widening && !swapping;
    if (widening) {
      la = ua;
      t_lo = t_up;
      ua = fminf(2.f * ua, MAX_STEP);
    }
    prev_error = err;
  }

  const float lower_error = E0 + t_lo * g0 + 0.5f * t_lo * t_lo * h;

  // zoom phase
  for (int z = 0; z < ZOOM_ITERS; ++z) {
    const float lg = g0 + t_lo * h;
    const float ug = g0 + t_up * h;
    const float diff = ug - lg;
    const float safe = (diff == 0.f) ? 1.f : diff + 1e-8f;
    float cand = la - lg * (ua - la) / safe;       // secant step
    const bool nonlin = (diff == 0.f) || (cand < fminf(la, ua)) || (cand > fmaxf(la, ua));
    if (nonlin) cand = 0.5f * (la + ua);
    const float cand_w = zooming ? cand : 0.f;
    const float t_cand = zooming ? cand_w : t_up;  // cand_x param
    const float err = E0 + t_cand * g0 + 0.5f * t_cand * t_cand * h;
    const bool inc = zooming &&
        ((err > E0 + WOLFE_C1 * cand_w * g0) || (err >= lower_error));
    if (inc) { ua = cand_w; t_up = t_cand; }
    const bool noninc = zooming && !inc;
    const float gdot = g0 + t_cand * h;
    const bool met = noninc && (fabsf(gdot) <= -WOLFE_C2 * g0);
    if (met) { ua = cand_w; t_up = t_cand; }
    zooming = zooming && !met;
    const bool swapping = noninc && (gdot * (ua - la) >= 0.f) && zooming;
    if (swapping) {
      float tt = ua; ua = la; la = tt;
      tt = t_up; t_up = t_lo; t_lo = tt;
    }
    const bool lowupd = noninc && zooming;         // includes swapped lanes
    if (lowupd) { la = cand_w; t_lo = t_cand; }
  }

  // write upper_x = x0 + t_up * sd (vectorized b128)
  const float4* __restrict__ x4 = reinterpret_cast<const float4*>(x0 + (size_t)i * D_DIM);
  const float4* __restrict__ s4 = reinterpret_cast<const float4*>(sd + (size_t)i * D_DIM);
  float4* __restrict__ o4 = reinterpret_cast<float4*>(out + (size_t)i * D_DIM);
  for (int q = threadIdx.x; q < D_DIM / 4; q += 256) {
    const float4 a = x4[q];
    const float4 s = s4[q];
    float4 r;
    r.x = __builtin_fmaf(t_up, s.x, a.x);
    r.y = __builtin_fmaf(t_up, s.y, a.y);
    r.z = __builtin_fmaf(t_up, s.z, a.z);
    r.w = __builtin_fmaf(t_up, s.w, a.w);
    o4[q] = r;
  }
}

extern "C" void kernel_launch(void* const* d_in, const int* in_sizes, int n_in,
                              void* d_out, int out_size, void* d_ws, size_t ws_size,
                              hipStream_t stream) {
  const float* x0 = (const float*)d_in[0];   // [32,32,2048]
  const float* A  = (const float*)d_in[1];   // [2048,2048]
  const float* b  = (const float*)d_in[2];   // [2048]
  const float* sd = (const float*)d_in[3];   // [32,32,2048]
  float* out = (float*)d_out;                // [32,32,2048]
  float* ws  = (float*)d_ws;                 // ~204 KB used

  dim3 g1(16, 16);
  ls_gemm_kernel<<<g1, 256, 0, stream>>>(x0, A, sd, ws);
  ls_rowstats_kernel<<<ROWS, 256, 0, stream>>>(x0, b, sd, ws);
  ls_wolfe_out_kernel<<<ROWS, 256, 0, stream>>>(x0, sd, ws, out);
}